// GCNModel_70626442215973
// MI455X (gfx1250) — compile-verified
//
#include <hip/hip_runtime.h>
#include <hip/hip_bf16.h>

// GCN layer, rank-1 structure:
//   deg -> norms -> edge scatter -> out[n,j] = h[n]*W[j] + bias[j]
// Memory-bound (~620 MB total traffic, ~27us at 23.3 TB/s). Final broadcast
// uses V_WMMA_F32_16X16X4_F32 (A: 16x4 with K=0 col = h, B rows = W, C = bias).

#define DIM_LATENT 512
#define RL_INV (1.0f / 20000.0f)

typedef __attribute__((ext_vector_type(2))) float v2f;
typedef __attribute__((ext_vector_type(8))) float v8f;

__global__ void zero_f32_kernel(float* __restrict__ p, int n) {
    int i = blockIdx.x * blockDim.x + threadIdx.x;
    if (i < n) p[i] = 0.0f;
}

__global__ void degree_kernel(const int* __restrict__ src,
                              const int* __restrict__ dst,
                              float* __restrict__ deg_s,
                              float* __restrict__ deg_d, int E) {
    int i = blockIdx.x * blockDim.x + threadIdx.x;
    if (i < E) {
        atomicAdd(deg_s + src[i], 1.0f);
        atomicAdd(deg_d + dst[i], 1.0f);
    }
}

// In-place: deg_s[i] -> sx[i] = (rl/20000)*rsqrt(max(out_deg,1))
//           deg_d[i] -> norm_dst[i] = rsqrt(max(in_deg,1))
__global__ void norm_kernel(const float* __restrict__ rl,
                            float* __restrict__ deg_s,
                            float* __restrict__ deg_d, int N) {
    int i = blockIdx.x * blockDim.x + threadIdx.x;
    if (i < N) {
        float ns = rsqrtf(fmaxf(deg_s[i], 1.0f));
        float nd = rsqrtf(fmaxf(deg_d[i], 1.0f));
        deg_s[i] = (rl[i] * RL_INV) * ns;
        deg_d[i] = nd;
    }
}

__global__ void scatter_kernel(const int* __restrict__ src,
                               const int* __restrict__ dst,
                               const float* __restrict__ sx,
                               float* __restrict__ agg, int E) {
    int i = blockIdx.x * blockDim.x + threadIdx.x;
    if (i < E) atomicAdd(agg + dst[i], sx[src[i]]);
}

// One wave handles a 16-node tile; 32 WMMAs cover the 512 latent dims.
// A (16x4 f32, 2 VGPRs): lanes 0-15 VGPR0 = K=0 column = h[m]; rest zero.
// B (4x16 f32, 2 VGPRs): every K row = W[j]  (only K=0 multiplies non-zero A).
// C/D (16x16 f32, 8 VGPRs): lane -> N = lane&15; VGPR r -> M = r (+8 for hi half).
__global__ void outer_wmma_kernel(const float* __restrict__ agg,
                                  const float* __restrict__ ndst,
                                  const float* __restrict__ W,
                                  const float* __restrict__ bias,
                                  float* __restrict__ out, int N) {
    int wave = threadIdx.x >> 5;
    int lane = threadIdx.x & 31;
    int tile = blockIdx.x * (blockDim.x >> 5) + wave;
    int node_base = tile * 16;
    if (node_base >= N) return;   // wave-uniform: EXEC stays all-ones past here

    int jn = lane & 15;

    v2f a;
    a.x = 0.0f;
    a.y = 0.0f;
    if (lane < 16) {
        int n = node_base + lane;
        a.x = agg[n] * ndst[n];   // h[n], lands in K=0 column
    }

    int m0 = (lane < 16) ? 0 : 8;
    float* orow = out + (size_t)node_base * DIM_LATENT;

    for (int jb = 0; jb < DIM_LATENT; jb += 16) {
        float w  = W[jb + jn];
        float bs = bias[jb + jn];
        v2f b;
        b.x = w;
        b.y = w;
        v8f c = {bs, bs, bs, bs, bs, bs, bs, bs};
        c = __builtin_amdgcn_wmma_f32_16x16x4_f32(
                /*neg_a=*/false, a, /*neg_b=*/false, b,
                /*c_mod=*/(short)0, c, /*reuse_a=*/false, /*reuse_b=*/false);
        float* p = orow + (size_t)m0 * DIM_LATENT + jb + jn;
#pragma unroll
        for (int r = 0; r < 8; ++r)
            p[(size_t)r * DIM_LATENT] = c[r];   // 2x 64B contiguous segments/store
    }
}

extern "C" void kernel_launch(void* const* d_in, const int* in_sizes, int n_in,
                              void* d_out, int out_size, void* d_ws, size_t ws_size,
                              hipStream_t stream) {
    const float* rl   = (const float*)d_in[0];  // read_length [N]
    const float* W    = (const float*)d_in[1];  // weight [1,512]
    const float* bias = (const float*)d_in[2];  // bias [512]
    const int*   ei   = (const int*)d_in[3];    // edge_index [2,E]

    const int N = in_sizes[0];
    const int E = in_sizes[3] / 2;
    const int* src = ei;
    const int* dst = ei + E;

    float* ws    = (float*)d_ws;
    float* deg_s = ws;          // becomes sx after norm_kernel
    float* deg_d = ws + N;      // becomes norm_dst after norm_kernel
    float* agg   = ws + 2 * (size_t)N;
    float* out   = (float*)d_out;

    const int T = 256;
    zero_f32_kernel<<<(3 * N + T - 1) / T, T, 0, stream>>>(ws, 3 * N);
    degree_kernel<<<(E + T - 1) / T, T, 0, stream>>>(src, dst, deg_s, deg_d, E);
    norm_kernel<<<(N + T - 1) / T, T, 0, stream>>>(rl, deg_s, deg_d, N);
    scatter_kernel<<<(E + T - 1) / T, T, 0, stream>>>(src, dst, deg_s, agg, E);

    int tiles = (N + 15) / 16;          // 12500 (N divisible by 16)
    int waves_per_block = T / 32;       // 8
    int blocks = (tiles + waves_per_block - 1) / waves_per_block;
    outer_wmma_kernel<<<blocks, T, 0, stream>>>(agg, deg_d, W, bias, out, N);
}